// LSTM_tele_lr_50233937494105
// MI455X (gfx1250) — compile-verified
//
#include <hip/hip_runtime.h>
#include <hip/hip_bf16.h>
#include <math.h>

typedef __attribute__((ext_vector_type(2))) float v2f;
typedef __attribute__((ext_vector_type(8))) float v8f;

#define D_IN   131072
#define H      512
#define G4H    2048      // 4*H
#define OUTDIM 128
#define NG     16384     // OUT*OUT

__device__ __forceinline__ float sigmoidf_(float x) {
    return 1.0f / (1.0f + __expf(-x));
}

// ---------------------------------------------------------------------------
// Kernel 1: partial[row] = W_ih1[row,:] . concat(dU, dV)   (1.07 GB stream)
// One block per row, coalesced float4 loads, LDS tree reduction.
// ---------------------------------------------------------------------------
__global__ void big_gemv_kernel(const float* __restrict__ W,
                                const float* __restrict__ dU,
                                const float* __restrict__ dV,
                                float* __restrict__ partial) {
    const int row = blockIdx.x;
    const float4* __restrict__ wr = (const float4*)(W + (size_t)row * D_IN);
    const float4* __restrict__ u4 = (const float4*)dU;
    const float4* __restrict__ v4 = (const float4*)dV;
    float acc = 0.0f;
    // 131072 floats = 32768 float4 chunks; boundary 16384 is a multiple of 256
    // so each iteration is branch-uniform per wave.
    for (int k = threadIdx.x; k < D_IN / 4; k += 256) {
        float4 w = wr[k];
        float4 x = (k < 16384) ? u4[k] : v4[k - 16384];
        acc = fmaf(w.x, x.x, acc);
        acc = fmaf(w.y, x.y, acc);
        acc = fmaf(w.z, x.z, acc);
        acc = fmaf(w.w, x.w, acc);
    }
    __shared__ float red[256];
    red[threadIdx.x] = acc;
    __syncthreads();
    for (int s = 128; s > 0; s >>= 1) {
        if (threadIdx.x < (unsigned)s) red[threadIdx.x] += red[threadIdx.x + s];
        __syncthreads();
    }
    if (threadIdx.x == 0) partial[row] = red[0];
}

// ---------------------------------------------------------------------------
// WMMA GEMV core: one wave computes y[rowBase..rowBase+15] = W[16 x K] . x
// A tile (16x4 f32):  lanes 0-15 -> row M=lane, K=0,1 ; lanes 16-31 -> K=2,3
// B tile (4x16 f32):  B(k,n) = x[k] broadcast across all 16 columns
// C/D (16x16 f32):    lane 0 VGPR v -> (M=v, N=0); lane 16 VGPR v -> (M=v+8, N=0)
// ---------------------------------------------------------------------------
__device__ __forceinline__ v8f wmma_gemv16(const float* __restrict__ W,
                                           int rowBase,
                                           const float* __restrict__ x,
                                           v8f acc, int lane) {
    const int m    = lane & 15;
    const int koff = (lane >> 4) << 1;            // 0 or 2
    const float* __restrict__ wrow = W + (size_t)(rowBase + m) * H + koff;
    const float* __restrict__ xv   = x + koff;
    #pragma unroll 4
    for (int k = 0; k < H; k += 4) {
        v2f a = *(const v2f*)(wrow + k);          // 8B aligned (koff even, k%4==0)
        v2f b = *(const v2f*)(xv + k);
        acc = __builtin_amdgcn_wmma_f32_16x16x4_f32(
            /*neg_a=*/false, a, /*neg_b=*/false, b,
            /*c_mod=*/(short)0, acc, /*reuse_a=*/false, /*reuse_b=*/false);
    }
    return acc;
}

// gates1[j] = partial[j] + b_ih[j] + b_hh[j] + W_hh1[j,:] . h_in
__global__ void lstm1_gates_kernel(const float* __restrict__ partial,
                                   const float* __restrict__ Whh,
                                   const float* __restrict__ h_in,
                                   const float* __restrict__ b_ih,
                                   const float* __restrict__ b_hh,
                                   float* __restrict__ gates) {
    const int lane    = threadIdx.x & 31;
    const int wave    = (blockIdx.x * blockDim.x + threadIdx.x) >> 5;
    const int rowBase = wave * 16;                // 128 waves * 16 = 2048 rows
    v8f acc = {};
    acc = wmma_gemv16(Whh, rowBase, h_in, acc, lane);
    if (lane == 0 || lane == 16) {
        const int mbase = rowBase + ((lane >> 4) << 3);
        #pragma unroll
        for (int v = 0; v < 8; ++v) {
            const int r = mbase + v;
            gates[r] = acc[v] + partial[r] + b_ih[r] + b_hh[r];
        }
    }
}

// gates2[j] = b_ih[j] + b_hh[j] + W_ih2[j,:] . h0  +  W_hh2[j,:] . h1_in
__global__ void lstm2_gates_kernel(const float* __restrict__ Wih,
                                   const float* __restrict__ Whh,
                                   const float* __restrict__ h0,
                                   const float* __restrict__ h1_in,
                                   const float* __restrict__ b_ih,
                                   const float* __restrict__ b_hh,
                                   float* __restrict__ gates) {
    const int lane    = threadIdx.x & 31;
    const int wave    = (blockIdx.x * blockDim.x + threadIdx.x) >> 5;
    const int rowBase = wave * 16;
    v8f acc = {};
    acc = wmma_gemv16(Wih, rowBase, h0,    acc, lane);
    acc = wmma_gemv16(Whh, rowBase, h1_in, acc, lane);
    if (lane == 0 || lane == 16) {
        const int mbase = rowBase + ((lane >> 4) << 3);
        #pragma unroll
        for (int v = 0; v < 8; ++v) {
            const int r = mbase + v;
            gates[r] = acc[v] + b_ih[r] + b_hh[r];
        }
    }
}

// c_new = sigmoid(f)*c + sigmoid(i)*tanh(g); h_new = sigmoid(o)*tanh(c_new)
__global__ void cell_update_kernel(const float* __restrict__ gates,
                                   const float* __restrict__ c_in,
                                   float* __restrict__ h_out,
                                   float* __restrict__ c_out) {
    const int j = blockIdx.x * blockDim.x + threadIdx.x;
    if (j < H) {
        const float ig = sigmoidf_(gates[j]);
        const float fg = sigmoidf_(gates[H + j]);
        const float gg = tanhf(gates[2 * H + j]);
        const float og = sigmoidf_(gates[3 * H + j]);
        const float c  = fg * c_in[j] + ig * gg;
        c_out[j] = c;
        h_out[j] = og * tanhf(c);
    }
}

// g[row] = W1[row,:] . h1 + b1[row]   (16384 rows, WMMA GEMV)
__global__ void head_kernel(const float* __restrict__ W1,
                            const float* __restrict__ b1,
                            const float* __restrict__ h1,
                            float* __restrict__ g) {
    const int lane    = threadIdx.x & 31;
    const int wave    = (blockIdx.x * blockDim.x + threadIdx.x) >> 5;
    const int rowBase = wave * 16;                // 1024 waves * 16 = 16384 rows
    v8f acc = {};
    acc = wmma_gemv16(W1, rowBase, h1, acc, lane);
    if (lane == 0 || lane == 16) {
        const int mbase = rowBase + ((lane >> 4) << 3);
        #pragma unroll
        for (int v = 0; v < 8; ++v) {
            const int r = mbase + v;
            g[r] = acc[v] + b1[r];
        }
    }
}

// step = clip(W2.h1 + b2, 1e-7, 0.005); m_c = clip(2*clip(W3.h0 + b3, 0,1), 0,1)
__global__ void scalars_kernel(const float* __restrict__ W2, const float* __restrict__ b2,
                               const float* __restrict__ W3, const float* __restrict__ b3,
                               const float* __restrict__ h1, const float* __restrict__ h0,
                               float* __restrict__ out_step, float* __restrict__ out_mc) {
    __shared__ float s2[256];
    __shared__ float s3[256];
    const int t = threadIdx.x;
    float a2 = 0.0f, a3 = 0.0f;
    for (int k = t; k < H; k += 256) {
        a2 = fmaf(W2[k], h1[k], a2);
        a3 = fmaf(W3[k], h0[k], a3);
    }
    s2[t] = a2; s3[t] = a3;
    __syncthreads();
    for (int s = 128; s > 0; s >>= 1) {
        if (t < s) { s2[t] += s2[t + s]; s3[t] += s3[t + s]; }
        __syncthreads();
    }
    if (t == 0) {
        float step = s2[0] + b2[0];
        step = fminf(fmaxf(step, 1e-7f), 0.005f);
        float mc = s3[0] + b3[0];
        mc = 2.0f * fminf(fmaxf(mc, 0.0f), 1.0f);
        mc = fminf(fmaxf(mc, 0.0f), 1.0f);
        *out_step = step;
        *out_mc   = mc;
    }
}

extern "C" void kernel_launch(void* const* d_in, const int* in_sizes, int n_in,
                              void* d_out, int out_size, void* d_ws, size_t ws_size,
                              hipStream_t stream) {
    const float* dU      = (const float*)d_in[0];
    const float* dV      = (const float*)d_in[1];
    const float* hidden  = (const float*)d_in[2];   // [2,1,512]
    const float* cellst  = (const float*)d_in[3];   // [2,1,512]
    const float* W_ih1   = (const float*)d_in[4];
    const float* W_hh1   = (const float*)d_in[5];
    const float* b_ih1   = (const float*)d_in[6];
    const float* b_hh1   = (const float*)d_in[7];
    const float* W_ih2   = (const float*)d_in[8];
    const float* W_hh2   = (const float*)d_in[9];
    const float* b_ih2   = (const float*)d_in[10];
    const float* b_hh2   = (const float*)d_in[11];
    const float* W1      = (const float*)d_in[12];
    const float* b1      = (const float*)d_in[13];
    const float* W2      = (const float*)d_in[14];
    const float* b2      = (const float*)d_in[15];
    const float* W3      = (const float*)d_in[16];
    const float* b3      = (const float*)d_in[17];

    float* out = (float*)d_out;
    // Output layout (flat, return order):
    float* g_out    = out;                 // [0, 16384)
    float* step_out = out + NG;            // 16384
    float* mc_out   = out + NG + 1;        // 16385
    float* h0_out   = out + NG + 2;        // h stack [0]
    float* h1_out   = out + NG + 2 + H;    // h stack [1]
    float* c0_out   = out + NG + 2 + 2*H;  // c stack [0]
    float* c1_out   = out + NG + 2 + 3*H;  // c stack [1]

    float* ws       = (float*)d_ws;
    float* partial1 = ws;                  // 2048
    float* gates1   = ws + G4H;            // 2048
    float* gates2   = ws + 2 * G4H;        // 2048

    // Layer 1: dominant 1.07 GB streaming GEMV
    big_gemv_kernel<<<G4H, 256, 0, stream>>>(W_ih1, dU, dV, partial1);
    // Layer 1 hidden part + biases (WMMA GEMV, 128 waves)
    lstm1_gates_kernel<<<32, 128, 0, stream>>>(partial1, W_hh1, hidden /*h[0]*/,
                                               b_ih1, b_hh1, gates1);
    cell_update_kernel<<<2, 256, 0, stream>>>(gates1, cellst /*c[0]*/, h0_out, c0_out);
    // Layer 2 (WMMA GEMV x2, 128 waves)
    lstm2_gates_kernel<<<32, 128, 0, stream>>>(W_ih2, W_hh2, h0_out, hidden + H /*h[1]*/,
                                               b_ih2, b_hh2, gates2);
    cell_update_kernel<<<2, 256, 0, stream>>>(gates2, cellst + H /*c[1]*/, h1_out, c1_out);
    // Output head (WMMA GEMV, 1024 waves)
    head_kernel<<<128, 256, 0, stream>>>(W1, b1, h1_out, g_out);
    // Scalar heads
    scalars_kernel<<<1, 256, 0, stream>>>(W2, b2, W3, b3, h1_out, h0_out, step_out, mc_out);
}